// NURBSLayer_12618613916415
// MI455X (gfx1250) — compile-verified
//
#include <hip/hip_runtime.h>

// Problem constants (from the reference)
#define N_CP    128
#define N_DATA  32768
#define BATCH   16
#define ROWS    64            // 4 rows per batch: 3 weighted-cp rows + 1 weight row
#define WAVES_PER_WG 8
#define THREADS (WAVES_PER_WG * 32)

typedef __attribute__((ext_vector_type(2))) float v2f;
typedef __attribute__((ext_vector_type(8))) float v8f;

// Write one 16x16 C tile: tile mt covers global rows [mt*16, mt*16+16) = batches
// 4mt..4mt+3. In the 32-bit C/D layout, lane-half hi holds rows j+8*hi (j = VGPR
// index 0..7): batch = 4mt + 2hi + (j>=4), d = j&3; divisor rows (d==3) are VGPR 3
// and VGPR 7 of the SAME lane -> no cross-lane ops for the NURBS rational divide.
__device__ __forceinline__ void write_tile(float* __restrict__ out, v8f a,
                                           int mt, int hi, int n) {
    float inv0 = 1.0f / a[3];
    float inv1 = 1.0f / a[7];
    int b0 = 4 * mt + 2 * hi;
    float* o0 = out + (size_t)(b0 * 3) * N_DATA + n;
    o0[0 * (size_t)N_DATA] = a[0] * inv0;
    o0[1 * (size_t)N_DATA] = a[1] * inv0;
    o0[2 * (size_t)N_DATA] = a[2] * inv0;
    float* o1 = out + (size_t)((b0 + 1) * 3) * N_DATA + n;
    o1[0 * (size_t)N_DATA] = a[4] * inv1;
    o1[1 * (size_t)N_DATA] = a[5] * inv1;
    o1[2 * (size_t)N_DATA] = a[6] * inv1;
}

__global__ __launch_bounds__(THREADS)
void nurbs_wmma_kernel(const float* __restrict__ cp,   // [16,3,128]
                       const float* __restrict__ w,    // [16,1,128]
                       const float* __restrict__ Nb,   // [128,32768] row-major
                       float* __restrict__ out)        // [16,3,32768]
{
    __shared__ float Alds[ROWS * N_CP];   // 32 KB of a 320 KB WGP LDS

    const int tid = threadIdx.x;

    // Stage A = [cp*w rows ; w row] into LDS (fused scaling, read once per WG).
    for (int idx = tid; idx < ROWS * N_CP; idx += THREADS) {
        int row = idx >> 7;         // /128
        int c   = idx & (N_CP - 1);
        int b   = row >> 2;
        int d   = row & 3;
        float wv = w[b * N_CP + c];
        Alds[idx] = (d == 3) ? wv : cp[(b * 3 + d) * N_CP + c] * wv;
    }
    __syncthreads();

    const int wave = tid >> 5;
    const int lane = tid & 31;
    const int lo   = lane & 15;     // column / M-row within tile
    const int hi   = lane >> 4;     // K sub-pair selector for A/B fragments

    // Each wave owns one 16-column strip of N / C.
    const int n0 = (blockIdx.x * WAVES_PER_WG + wave) * 16;
    const float* bcol = Nb + n0 + lo;

    v8f acc0 = {}, acc1 = {}, acc2 = {}, acc3 = {};

    #pragma unroll 4
    for (int k0 = 0; k0 < N_CP; k0 += 4) {
        const int krow = k0 + 2 * hi;

        // B fragment (4x16 fp32, K x N): VGPR0 = row krow, VGPR1 = row krow+1.
        v2f bf;
        bf.x = bcol[(size_t)krow * N_DATA];
        bf.y = bcol[(size_t)(krow + 1) * N_DATA];

        // Prefetch one full unrolled body (4 K-steps = 16 rows) ahead so the
        // prefetch actually leads the demand loads; locality 3 -> WGP scope
        // (pull into all cache levels), not the useless SYS-scope GL2-only hint.
        if (k0 + 16 < N_CP) {
            __builtin_prefetch(&bcol[(size_t)(krow + 16) * N_DATA], 0, 3);
            __builtin_prefetch(&bcol[(size_t)(krow + 17) * N_DATA], 0, 3);
        }

        // A fragments (16x4 fp32): lane lo = row M, VGPRs hold K = krow, krow+1.
        // Consecutive fp32 in LDS row-major A -> ds_load_2addr_stride64_b64 pairs.
        const v2f a0 = *(const v2f*)&Alds[( 0 + lo) * N_CP + krow];
        const v2f a1 = *(const v2f*)&Alds[(16 + lo) * N_CP + krow];
        const v2f a2 = *(const v2f*)&Alds[(32 + lo) * N_CP + krow];
        const v2f a3 = *(const v2f*)&Alds[(48 + lo) * N_CP + krow];

        acc0 = __builtin_amdgcn_wmma_f32_16x16x4_f32(false, a0, false, bf, (short)0, acc0, false, false);
        acc1 = __builtin_amdgcn_wmma_f32_16x16x4_f32(false, a1, false, bf, (short)0, acc1, false, false);
        acc2 = __builtin_amdgcn_wmma_f32_16x16x4_f32(false, a2, false, bf, (short)0, acc2, false, false);
        acc3 = __builtin_amdgcn_wmma_f32_16x16x4_f32(false, a3, false, bf, (short)0, acc3, false, false);
    }

    const int n = n0 + lo;
    write_tile(out, acc0, 0, hi, n);
    write_tile(out, acc1, 1, hi, n);
    write_tile(out, acc2, 2, hi, n);
    write_tile(out, acc3, 3, hi, n);
}

extern "C" void kernel_launch(void* const* d_in, const int* in_sizes, int n_in,
                              void* d_out, int out_size, void* d_ws, size_t ws_size,
                              hipStream_t stream) {
    // setup_inputs order: input (unused), control_points, weights, N
    const float* cp = (const float*)d_in[1];   // [16,3,128]
    const float* w  = (const float*)d_in[2];   // [16,1,128]
    const float* Nb = (const float*)d_in[3];   // [128,32768]
    float* out = (float*)d_out;                // [16,3,32768]

    const int strips = N_DATA / 16;                 // 2048
    const int blocks = strips / WAVES_PER_WG;       // 256
    nurbs_wmma_kernel<<<blocks, THREADS, 0, stream>>>(cp, w, Nb, out);
}